// MagPhaseLoss_4853313045063
// MI455X (gfx1250) — compile-verified
//
#include <hip/hip_runtime.h>
#include <math.h>

#define B_SZ    16
#define SAMPLES 524288
#define N_FFT   1024
#define HOP     256
#define FBINS   513
#define FRAMES  2049
#define NEL     (B_SZ * FBINS * FRAMES)   /* 16,818,192 */

typedef __attribute__((ext_vector_type(2))) float v2f;
typedef __attribute__((ext_vector_type(8))) float v8f;

__device__ __forceinline__ float wrap_abs(float d) {
    const float TWO_PI = 6.28318530717958647692f;
    return fabsf(d - TWO_PI * rintf(d * (1.0f / TWO_PI)));
}

__global__ void init_acc_kernel(float* acc) {
    if (threadIdx.x < 2) acc[threadIdx.x] = 0.f;
}

// One block = one (batch, frame) column of the STFT.
// 16 parallel FFT-64 in LDS, twiddle, then the final 16-point DFT across
// sub-sequences as a complex GEMM on the matrix core (v_wmma_f32_16x16x4_f32).
__global__ __launch_bounds__(256) void stft_kernel(
        const float* __restrict__ gt, const float* __restrict__ win,
        float* __restrict__ lm_ws, float* __restrict__ ph_ws) {
    __shared__ float s_xr[1024], s_xi[1024], s_or[1024], s_oi[1024];
    __shared__ float s_twc[1024], s_tws[1024];     // omega_1024^t, t=0..1023
    const int f = blockIdx.x;
    const int b = blockIdx.y;
    const int tid = threadIdx.x;
    const float NEG2PI = -6.28318530717958647692f;

    // ---- full twiddle table: 4 sincos per thread, branchless lookups later ----
    for (int t = tid; t < 1024; t += 256) {
        float c, sn;
        __sincosf(NEG2PI * (float)t * (1.0f / 1024.0f), &sn, &c);
        s_twc[t] = c;
        s_tws[t] = sn;
    }

    // ---- load + Hann window + reflect pad; row n1 = n%16, col = bitrev6(n/16) ----
    for (int n = tid; n < 1024; n += 256) {
        int idx = f * HOP + n - 512;
        if (idx < 0) idx = -idx;
        if (idx >= SAMPLES) idx = 2 * (SAMPLES - 1) - idx;
        float v = gt[(size_t)b * SAMPLES + idx] * win[n];
        int n1 = n & 15;
        int rb = __brev((unsigned)(n >> 4)) >> 26;
        s_xr[n1 * 64 + rb] = v;
        s_xi[n1 * 64 + rb] = 0.f;
    }
    __syncthreads();

    // ---- 16 independent radix-2 DIT FFT-64 (512 butterflies / stage) ----
    // stage twiddle: omega_m^pos = omega_1024^(pos << (10-s))
    for (int s = 1; s <= 6; ++s) {
        int m = 1 << s, half = m >> 1;
        for (int i = tid; i < 512; i += 256) {
            int row = i >> 5;
            int bf  = i & 31;
            int grp = bf / half;
            int pos = bf & (half - 1);
            int base = row * 64 + grp * m + pos;
            int e = pos << (10 - s);
            float c = s_twc[e], sn = s_tws[e];
            float ar = s_xr[base],        ai = s_xi[base];
            float br = s_xr[base + half], bi = s_xi[base + half];
            float tr = c * br - sn * bi;
            float ti = c * bi + sn * br;
            s_xr[base]        = ar + tr;
            s_xi[base]        = ai + ti;
            s_xr[base + half] = ar - tr;
            s_xi[base + half] = ai - ti;
        }
        __syncthreads();
    }

    // ---- inter-stage twiddle: Z[n1,k2] = Y[n1,k2] * omega_1024^(n1*k2) ----
    for (int i = tid; i < 1024; i += 256) {
        int n1 = i >> 6, k2 = i & 63;
        int e = n1 * k2;                         // 0..945
        float c = s_twc[e], sn = s_tws[e];
        float yr = s_xr[i], yi = s_xi[i];
        s_xr[i] = c * yr - sn * yi;
        s_xi[i] = c * yi + sn * yr;
    }
    __syncthreads();

    // ---- X[64*k1 + k2] = sum_n1 F16[k1,n1] * Z[n1,k2] via WMMA f32 16x16x4 ----
    // F16[k1,n1] = omega_16^(k1*n1) = omega_1024^(64*((k1*n1)&15))
    // 8 waves: waves 0-3 -> real part of N-tiles 0-3, waves 4-7 -> imag part.
    {
        int wv   = tid >> 5;
        int lane = tid & 31;
        int tile = wv & 3;       // k2 range [16*tile, 16*tile+16)
        int part = wv >> 2;      // 0: Xr = Fr*Zr - Fi*Zi ; 1: Xi = Fr*Zi + Fi*Zr
        int mn   = lane & 15;    // A: M (=k1) ; B: N (=k2 local)
        int hi   = lane >> 4;
        const float sFi = part ? 1.0f : -1.0f;
        const float* Bp0 = part ? s_xi : s_xr;
        const float* Bp1 = part ? s_xr : s_xi;
        v8f acc = {};
        #pragma unroll
        for (int j = 0; j < 4; ++j) {   // product 0: A = Fr (cos)
            int n1a = 4 * j + 2 * hi, n1b = n1a + 1;
            int ea = ((mn * n1a) & 15) << 6;
            int eb = ((mn * n1b) & 15) << 6;
            v2f a, bm;
            a.x = s_twc[ea];
            a.y = s_twc[eb];
            bm.x = Bp0[n1a * 64 + 16 * tile + mn];
            bm.y = Bp0[n1b * 64 + 16 * tile + mn];
            acc = __builtin_amdgcn_wmma_f32_16x16x4_f32(false, a, false, bm,
                                                        (short)0, acc, false, false);
        }
        #pragma unroll
        for (int j = 0; j < 4; ++j) {   // product 1: A = +/-Fi (sin)
            int n1a = 4 * j + 2 * hi, n1b = n1a + 1;
            int ea = ((mn * n1a) & 15) << 6;
            int eb = ((mn * n1b) & 15) << 6;
            v2f a, bm;
            a.x = sFi * s_tws[ea];
            a.y = sFi * s_tws[eb];
            bm.x = Bp1[n1a * 64 + 16 * tile + mn];
            bm.y = Bp1[n1b * 64 + 16 * tile + mn];
            acc = __builtin_amdgcn_wmma_f32_16x16x4_f32(false, a, false, bm,
                                                        (short)0, acc, false, false);
        }
        float* od = part ? s_oi : s_or;
        #pragma unroll
        for (int v = 0; v < 8; ++v) {   // C/D layout: vgpr v, lane-half hi -> M = v + 8*hi
            int k1 = v + 8 * hi;
            od[64 * k1 + 16 * tile + mn] = acc[v];
        }
    }
    __syncthreads();

    // ---- bins 0..512: log|X| and angle(X), native [b][k][f] layout ----
    for (int k = tid; k < FBINS; k += 256) {
        float re = s_or[k], im = s_oi[k];
        float mag = sqrtf(re * re + im * im);
        size_t o = ((size_t)b * FBINS + k) * (size_t)FRAMES + f;
        lm_ws[o] = logf(mag + 1e-14f);
        ph_ws[o] = atan2f(im, re);
    }
}

// All four loss terms; frame index fastest -> fully coalesced global reads.
__global__ __launch_bounds__(256) void loss_kernel(
        const float* __restrict__ pmag, const float* __restrict__ pph,
        const float* __restrict__ lm_ws, const float* __restrict__ ph_ws,
        float* __restrict__ acc) {
    const float LNB = 0.0017896303f;   // ln(2.5)/512
    float msum = 0.f, psum = 0.f;
    for (int i = blockIdx.x * blockDim.x + threadIdx.x; i < NEL;
         i += gridDim.x * blockDim.x) {
        int t = i % FRAMES;
        int k = (i / FRAMES) % FBINS;
        float w  = __expf(LNB * (float)k);
        float tp = ph_ws[i], pv = pph[i];
        msum += fabsf(pmag[i] - lm_ws[i]);
        float d0 = pv - tp;
        psum += w * wrap_abs(d0);
        float dfd = (k == 0) ? -d0
                             : ((pph[i - FRAMES] - pv) - (ph_ws[i - FRAMES] - tp));
        psum += w * wrap_abs(dfd);
        float dtd = (t == 0) ? -d0
                             : ((pph[i - 1] - pv) - (ph_ws[i - 1] - tp));
        psum += w * wrap_abs(dtd);
    }
    __shared__ float rm[256], rp[256];
    int tid = threadIdx.x;
    rm[tid] = msum; rp[tid] = psum;
    __syncthreads();
    for (int s = 128; s > 0; s >>= 1) {
        if (tid < s) { rm[tid] += rm[tid + s]; rp[tid] += rp[tid + s]; }
        __syncthreads();
    }
    if (tid == 0) {
        atomicAdd(&acc[0], rm[0]);
        atomicAdd(&acc[1], rp[0]);
    }
}

__global__ void finalize_kernel(const float* __restrict__ acc,
                                float* __restrict__ out) {
    if (threadIdx.x == 0) {
        const float inv = 1.0f / (float)NEL;
        out[0] = acc[0] * inv;   // mag_loss
        out[1] = acc[1] * inv;   // phase_loss (sum of three equal-count means)
    }
}

extern "C" void kernel_launch(void* const* d_in, const int* in_sizes, int n_in,
                              void* d_out, int out_size, void* d_ws, size_t ws_size,
                              hipStream_t stream) {
    const float* pmag = (const float*)d_in[0];   // pred_magnitude (B,FBINS,FRAMES)
    const float* pph  = (const float*)d_in[1];   // pred_phase
    const float* gt   = (const float*)d_in[2];   // (B, SAMPLES)
    const float* win  = (const float*)d_in[3];   // (1024,)
    (void)in_sizes; (void)n_in; (void)out_size; (void)ws_size;

    float* wsf   = (float*)d_ws;
    float* lm_ws = wsf + 64;
    float* ph_ws = lm_ws + (size_t)NEL;
    float* out   = (float*)d_out;

    init_acc_kernel<<<1, 32, 0, stream>>>(wsf);
    stft_kernel<<<dim3(FRAMES, B_SZ), 256, 0, stream>>>(gt, win, lm_ws, ph_ws);
    loss_kernel<<<4096, 256, 0, stream>>>(pmag, pph, lm_ws, ph_ws, wsf);
    finalize_kernel<<<1, 32, 0, stream>>>(wsf, out);
}